// ResidualGCNBlock_60447369724690
// MI455X (gfx1250) — compile-verified
//
#include <hip/hip_runtime.h>

#define CIN  256
#define COUT 128
#define BN_EPS 1e-5f

typedef __attribute__((ext_vector_type(2))) float v2f;
typedef __attribute__((ext_vector_type(4))) float v4f;
typedef __attribute__((ext_vector_type(8))) float v8f;

// ---------------------------------------------------------------------------
// 0) Workspace init: h = 0, deg = 1.0 (self-loop weight), column sums = 0
// ---------------------------------------------------------------------------
__global__ void rgcn_init_ws(float* __restrict__ h, float* __restrict__ deg,
                             float* __restrict__ sums, long long totalHF, long long N) {
  long long idx = (long long)blockIdx.x * blockDim.x + threadIdx.x;
  if (idx < totalHF) h[idx] = 0.0f;
  if (idx < N)       deg[idx] = 1.0f;          // self loop contributes weight 1
  if (idx < 2 * COUT) sums[idx] = 0.0f;
}

// ---------------------------------------------------------------------------
// 1) Weighted in-degree:  deg[col] += ew[e]
// ---------------------------------------------------------------------------
__global__ void rgcn_degree(const long long* __restrict__ ei,
                            const float* __restrict__ ew,
                            float* __restrict__ deg, long long E) {
  long long e = (long long)blockIdx.x * blockDim.x + threadIdx.x;
  if (e < E) {
    long long c = ei[E + e];                   // target (col)
    atomicAdd(deg + c, ew[e]);
  }
}

// ---------------------------------------------------------------------------
// 2) dinv = deg > 0 ? rsqrt(deg) : 0   (in place)
// ---------------------------------------------------------------------------
__global__ void rgcn_dinv(float* __restrict__ deg, long long N) {
  long long i = (long long)blockIdx.x * blockDim.x + threadIdx.x;
  if (i < N) {
    float d = deg[i];
    deg[i] = (d > 0.0f) ? rsqrtf(d) : 0.0f;
  }
}

// ---------------------------------------------------------------------------
// 3) Dual fp32 WMMA GEMM:  xw = x @ W,  xr = x @ Ws
//    One wave computes a 16x64 slab for BOTH weight matrices (8 accumulators)
//    so each A fragment feeds 8 V_WMMA_F32_16X16X4_F32 ops.
// ---------------------------------------------------------------------------
__global__ __launch_bounds__(128)
void rgcn_gemm_dual_wmma(const float* __restrict__ x,
                         const float* __restrict__ W,
                         const float* __restrict__ Ws,
                         float* __restrict__ xw,
                         float* __restrict__ xr,
                         int Ntiles, long long N) {
  const int lane    = threadIdx.x & 31;
  const int wave    = threadIdx.x >> 5;
  const int rowTile = blockIdx.x * 4 + wave;          // wave-uniform guard
  if (rowTile >= Ntiles) return;                      // whole wave exits together

  const int halfSel = lane >> 4;                      // 0: lanes 0-15, 1: lanes 16-31
  const int l15     = lane & 15;
  long long m = (long long)rowTile * 16 + l15;        // A row this lane feeds
  if (m >= N) m = N - 1;                              // clamp (reads only)
  const int n0   = blockIdx.y * 64 + l15;             // base C column this lane holds
  const int koff = halfSel * 2;                       // K sub-offset per ISA A layout

  const float* xrow = x + m * CIN;

  v8f accW0 = {}, accW1 = {}, accW2 = {}, accW3 = {};
  v8f accS0 = {}, accS1 = {}, accS2 = {}, accS3 = {};

#pragma unroll 2
  for (int k0 = 0; k0 < CIN; k0 += 4) {
    // A fragment: lane holds x[m][k0+koff], x[m][k0+koff+1]  -> b64 load
    v2f a = *reinterpret_cast<const v2f*>(xrow + k0 + koff);

    const float* wr = W  + (long long)(k0 + koff) * COUT + n0;
    const float* zr = Ws + (long long)(k0 + koff) * COUT + n0;

    v2f bw0, bw1, bw2, bw3, bz0, bz1, bz2, bz3;
    bw0.x = wr[0];  bw0.y = wr[COUT +  0];
    bw1.x = wr[16]; bw1.y = wr[COUT + 16];
    bw2.x = wr[32]; bw2.y = wr[COUT + 32];
    bw3.x = wr[48]; bw3.y = wr[COUT + 48];
    bz0.x = zr[0];  bz0.y = zr[COUT +  0];
    bz1.x = zr[16]; bz1.y = zr[COUT + 16];
    bz2.x = zr[32]; bz2.y = zr[COUT + 32];
    bz3.x = zr[48]; bz3.y = zr[COUT + 48];

    accW0 = __builtin_amdgcn_wmma_f32_16x16x4_f32(false, a, false, bw0, (short)0, accW0, false, false);
    accW1 = __builtin_amdgcn_wmma_f32_16x16x4_f32(false, a, false, bw1, (short)0, accW1, false, false);
    accW2 = __builtin_amdgcn_wmma_f32_16x16x4_f32(false, a, false, bw2, (short)0, accW2, false, false);
    accW3 = __builtin_amdgcn_wmma_f32_16x16x4_f32(false, a, false, bw3, (short)0, accW3, false, false);
    accS0 = __builtin_amdgcn_wmma_f32_16x16x4_f32(false, a, false, bz0, (short)0, accS0, false, false);
    accS1 = __builtin_amdgcn_wmma_f32_16x16x4_f32(false, a, false, bz1, (short)0, accS1, false, false);
    accS2 = __builtin_amdgcn_wmma_f32_16x16x4_f32(false, a, false, bz2, (short)0, accS2, false, false);
    accS3 = __builtin_amdgcn_wmma_f32_16x16x4_f32(false, a, false, bz3, (short)0, accS3, false, false);
  }

  // D layout: VGPR v holds row (v + 8*halfSel) of the tile, column n
  const bool fullTile = ((long long)rowTile * 16 + 16) <= N;   // wave-uniform
  if (fullTile) {
#pragma unroll
    for (int v = 0; v < 8; ++v) {
      long long base = ((long long)rowTile * 16 + v + halfSel * 8) * COUT + n0;
      xw[base +  0] = accW0[v];
      xw[base + 16] = accW1[v];
      xw[base + 32] = accW2[v];
      xw[base + 48] = accW3[v];
      xr[base +  0] = accS0[v];
      xr[base + 16] = accS1[v];
      xr[base + 32] = accS2[v];
      xr[base + 48] = accS3[v];
    }
  } else {
#pragma unroll
    for (int v = 0; v < 8; ++v) {
      long long row = (long long)rowTile * 16 + v + halfSel * 8;
      if (row < N) {
        long long base = row * COUT + n0;
        xw[base +  0] = accW0[v];
        xw[base + 16] = accW1[v];
        xw[base + 32] = accW2[v];
        xw[base + 48] = accW3[v];
        xr[base +  0] = accS0[v];
        xr[base + 16] = accS1[v];
        xr[base + 32] = accS2[v];
        xr[base + 48] = accS3[v];
      }
    }
  }
}

// ---------------------------------------------------------------------------
// 4) Edge scatter: one wave32 per (edge or self-loop), 4 floats per lane.
//    h[col] += dinv[row]*w*dinv[col] * xw[row]   via GLOBAL_ATOMIC_ADD_F32
// ---------------------------------------------------------------------------
__global__ __launch_bounds__(256)
void rgcn_scatter(const long long* __restrict__ ei,
                  const float* __restrict__ ew,
                  const float* __restrict__ dinv,
                  const float* __restrict__ xw,
                  float* __restrict__ h,
                  long long E, long long N) {
  long long gw = ((long long)blockIdx.x * blockDim.x + threadIdx.x) >> 5;
  const int lane = threadIdx.x & 31;
  const long long total = E + N;
  if (gw >= total) return;

  long long r, c;
  float w;
  if (gw < E) {
    r = ei[gw];            // source
    c = ei[E + gw];        // target
    w = ew[gw];
  } else {                 // self loop, weight 1
    r = gw - E;
    c = r;
    w = 1.0f;
  }
  const float norm = dinv[r] * w * dinv[c];

  const float* src = xw + r * COUT + lane * 4;
  float*       dst = h  + c * COUT + lane * 4;
  v4f v = *reinterpret_cast<const v4f*>(src);    // b128 gather (L2-resident)
  atomicAdd(dst + 0, norm * v.x);
  atomicAdd(dst + 1, norm * v.y);
  atomicAdd(dst + 2, norm * v.z);
  atomicAdd(dst + 3, norm * v.w);
}

// ---------------------------------------------------------------------------
// 5) BN statistics: per-column sum and sum-of-squares of (h + b)
// ---------------------------------------------------------------------------
__global__ __launch_bounds__(COUT)
void rgcn_bn_stats(const float* __restrict__ h, const float* __restrict__ b,
                   float* __restrict__ sums, long long N) {
  const int c = threadIdx.x;
  const float bc = b[c];
  float s = 0.0f, s2 = 0.0f;
  for (long long r = blockIdx.x; r < N; r += gridDim.x) {
    float v = h[r * COUT + c] + bc;
    s  += v;
    s2 += v * v;
  }
  atomicAdd(sums + c, s);
  atomicAdd(sums + COUT + c, s2);
}

// ---------------------------------------------------------------------------
// 6) Finalize:  out = relu((h+b-mean)*rsqrt(var+eps)*gamma + beta) + xr + bs
// ---------------------------------------------------------------------------
__global__ __launch_bounds__(256)
void rgcn_finalize(const float* __restrict__ h, const float* __restrict__ xr,
                   const float* __restrict__ b, const float* __restrict__ gamma,
                   const float* __restrict__ beta, const float* __restrict__ bs,
                   const float* __restrict__ sums, float* __restrict__ out,
                   long long total, float invN) {
  long long idx = (long long)blockIdx.x * blockDim.x + threadIdx.x;
  if (idx >= total) return;
  const int c = (int)(idx & (COUT - 1));
  const float mean = sums[c] * invN;
  const float var  = sums[COUT + c] * invN - mean * mean;
  const float val  = (h[idx] + b[c] - mean) * rsqrtf(var + BN_EPS) * gamma[c] + beta[c];
  out[idx] = fmaxf(val, 0.0f) + xr[idx] + bs[c];
}

// ---------------------------------------------------------------------------
extern "C" void kernel_launch(void* const* d_in, const int* in_sizes, int n_in,
                              void* d_out, int out_size, void* d_ws, size_t ws_size,
                              hipStream_t stream) {
  const float*     x     = (const float*)d_in[0];
  const long long* ei    = (const long long*)d_in[1];
  const float*     ew    = (const float*)d_in[2];
  const float*     W     = (const float*)d_in[3];
  const float*     b     = (const float*)d_in[4];
  const float*     gamma = (const float*)d_in[5];
  const float*     beta  = (const float*)d_in[6];
  const float*     Ws    = (const float*)d_in[7];
  const float*     bs    = (const float*)d_in[8];

  const long long N = (long long)in_sizes[0] / CIN;
  const long long E = (long long)in_sizes[2];

  // Workspace layout (floats): xw | xr | h | deg/dinv | sums(2*COUT)
  float* ws   = (float*)d_ws;
  float* xw   = ws;
  float* xr   = xw + N * COUT;
  float* h    = xr + N * COUT;
  float* deg  = h  + N * COUT;
  float* sums = deg + N;

  const long long totalHF = N * COUT;

  rgcn_init_ws<<<(unsigned)((totalHF + 255) / 256), 256, 0, stream>>>(h, deg, sums, totalHF, N);
  rgcn_degree <<<(unsigned)((E + 255) / 256), 256, 0, stream>>>(ei, ew, deg, E);
  rgcn_dinv   <<<(unsigned)((N + 255) / 256), 256, 0, stream>>>(deg, N);

  const int Ntiles = (int)((N + 15) / 16);
  dim3 ggrid((unsigned)((Ntiles + 3) / 4), COUT / 64);   // 4 waves/block, 16x64 slab/wave
  rgcn_gemm_dual_wmma<<<ggrid, 128, 0, stream>>>(x, W, Ws, xw, xr, Ntiles, N);

  const long long waves = E + N;                     // one wave per edge/self-loop
  rgcn_scatter<<<(unsigned)((waves * 32 + 255) / 256), 256, 0, stream>>>(ei, ew, deg, xw, h, E, N);

  rgcn_bn_stats<<<512, COUT, 0, stream>>>(h, b, sums, N);

  rgcn_finalize<<<(unsigned)((totalHF + 255) / 256), 256, 0, stream>>>(
      h, xr, b, gamma, beta, bs, sums, (float*)d_out, totalHF, 1.0f / (float)N);
}